// PSampling_22574348108041
// MI455X (gfx1250) — compile-verified
//
#include <hip/hip_runtime.h>
#include <hip/hip_bf16.h>
#include <math.h>

#define DDIM 128

typedef float v2f __attribute__((ext_vector_type(2)));
typedef float v8f __attribute__((ext_vector_type(8)));

// ---------------------------------------------------------------------------
// agg = (1 + eps[layer]) * h      (fused GIN self-term, saves a full pass)
// ---------------------------------------------------------------------------
__global__ void scatter_init_kernel(const float* __restrict__ h,
                                    float* __restrict__ agg,
                                    const float* __restrict__ eps_arr, int layer,
                                    long n) {
    long i = (long)blockIdx.x * blockDim.x + threadIdx.x;
    if (i < n) {
        float e = 1.0f + eps_arr[layer];
        agg[i] = e * h[i];
    }
}

// ---------------------------------------------------------------------------
// agg[dst[e]] += h[src[e]]   — one wave per edge, float4 per lane, f32 atomics
// ---------------------------------------------------------------------------
__global__ void edge_scatter_kernel(const float* __restrict__ h,
                                    const int* __restrict__ src,
                                    const int* __restrict__ dst,
                                    float* __restrict__ agg, int n_edges) {
    long tid = (long)blockIdx.x * blockDim.x + threadIdx.x;
    int e = (int)(tid >> 5);
    if (e >= n_edges) return;
    int lane = (int)(tid & 31);
    int s = src[e];
    int d = dst[e];
    const float4 v = *(const float4*)(h + (long)s * DDIM + lane * 4);
    float* p = agg + (long)d * DDIM + lane * 4;
    atomicAdd(p + 0, v.x);
    atomicAdd(p + 1, v.y);
    atomicAdd(p + 2, v.z);
    atomicAdd(p + 3, v.w);
}

// ---------------------------------------------------------------------------
// C = act(A @ W + bias),  A: n_rows x 128 (row-major), W: 128 x 128 (row-major)
// One wave per 16x16 C tile using V_WMMA_F32_16X16X4_F32, K-loop of 32 steps.
// A frag (16x4 f32, 2 VGPR): lane half selects K pair {0,1}/{2,3}.
// B frag (4x16 f32, 2 VGPR): same K split, col = lane & 15.
// C/D (8 VGPR): row = r + 8*(lane>>4), col = lane & 15.
// ---------------------------------------------------------------------------
__global__ void gemm_bias_act_kernel(const float* __restrict__ A,
                                     const float* __restrict__ W,
                                     const float* __restrict__ bias,
                                     float* __restrict__ C,
                                     int n_rows, int act) {
    const int waves_per_block = blockDim.x >> 5;
    int wave   = blockIdx.x * waves_per_block + (threadIdx.x >> 5);
    const int tiles_n = DDIM / 16;               // 8 column tiles
    int tile_m = wave / tiles_n;
    int tile_n = wave % tiles_n;
    if (tile_m * 16 >= n_rows) return;           // wave-uniform (exact grid: never taken)

    int lane = threadIdx.x & 31;
    int m    = lane & 15;
    int half = lane >> 4;

    const float* arow  = A + (long)(tile_m * 16 + m) * DDIM;
    const float* wbase = W + tile_n * 16 + (lane & 15);

    v8f acc = {};
#pragma unroll
    for (int k = 0; k < DDIM; k += 4) {
        v2f a, b;
        a.x = arow[k + 2 * half + 0];
        a.y = arow[k + 2 * half + 1];
        b.x = wbase[(long)(k + 2 * half + 0) * DDIM];
        b.y = wbase[(long)(k + 2 * half + 1) * DDIM];
        acc = __builtin_amdgcn_wmma_f32_16x16x4_f32(
            /*neg_a=*/false, a, /*neg_b=*/false, b,
            /*c_mod=*/(short)0, acc, /*reuse_a=*/false, /*reuse_b=*/false);
    }

    int col = tile_n * 16 + (lane & 15);
    float bv = bias[col];
#pragma unroll
    for (int r = 0; r < 8; ++r) {
        int row = tile_m * 16 + half * 8 + r;
        float v = acc[r] + bv;
        if (act) v = 1.0f / (1.0f + expf(-v));   // jax.nn.sigmoid
        C[(long)row * DDIM + col] = v;
    }
}

// ---------------------------------------------------------------------------
// Row-wise layernorm over D=128, one wave per row, butterfly reductions.
// ---------------------------------------------------------------------------
__global__ void layernorm_kernel(const float* __restrict__ in,
                                 const float* __restrict__ gamma,
                                 const float* __restrict__ beta,
                                 float* __restrict__ out, int n_rows) {
    int wave = blockIdx.x * (blockDim.x >> 5) + (threadIdx.x >> 5);
    if (wave >= n_rows) return;
    int lane = threadIdx.x & 31;

    const float4 v = ((const float4*)(in + (long)wave * DDIM))[lane];
    float s = v.x + v.y + v.z + v.w;
#pragma unroll
    for (int off = 16; off > 0; off >>= 1) s += __shfl_xor(s, off, 32);
    float mean = s * (1.0f / DDIM);

    float dx = v.x - mean, dy = v.y - mean, dz = v.z - mean, dw = v.w - mean;
    float q = dx * dx + dy * dy + dz * dz + dw * dw;
#pragma unroll
    for (int off = 16; off > 0; off >>= 1) q += __shfl_xor(q, off, 32);
    float rs = rsqrtf(q * (1.0f / DDIM) + 1e-5f);

    float4 g  = ((const float4*)gamma)[lane];
    float4 bb = ((const float4*)beta)[lane];
    float4 o;
    o.x = dx * rs * g.x + bb.x;
    o.y = dy * rs * g.y + bb.y;
    o.z = dz * rs * g.z + bb.z;
    o.w = dw * rs * g.w + bb.w;
    ((float4*)(out + (long)wave * DDIM))[lane] = o;
}

// ---------------------------------------------------------------------------
// JAX threefry2x32 (key(42) => (0,42)) + uniform->normal, bit-faithful to
// jax.random.normal(key(42), (N, D), float32).
// ---------------------------------------------------------------------------
__device__ __forceinline__ unsigned rotl32(unsigned x, int r) {
    return (x << r) | (x >> (32 - r));
}

__device__ __forceinline__ void threefry2x32(unsigned k0, unsigned k1,
                                             unsigned c0, unsigned c1,
                                             unsigned& o0, unsigned& o1) {
    unsigned ks0 = k0, ks1 = k1, ks2 = k0 ^ k1 ^ 0x1BD11BDAu;
    unsigned x0 = c0 + ks0, x1 = c1 + ks1;
    const int rotA[4] = {13, 15, 26, 6};
    const int rotB[4] = {17, 29, 16, 24};
#pragma unroll
    for (int i = 0; i < 4; ++i) { x0 += x1; x1 = rotl32(x1, rotA[i]); x1 ^= x0; }
    x0 += ks1; x1 += ks2 + 1u;
#pragma unroll
    for (int i = 0; i < 4; ++i) { x0 += x1; x1 = rotl32(x1, rotB[i]); x1 ^= x0; }
    x0 += ks2; x1 += ks0 + 2u;
#pragma unroll
    for (int i = 0; i < 4; ++i) { x0 += x1; x1 = rotl32(x1, rotA[i]); x1 ^= x0; }
    x0 += ks0; x1 += ks1 + 3u;
#pragma unroll
    for (int i = 0; i < 4; ++i) { x0 += x1; x1 = rotl32(x1, rotB[i]); x1 ^= x0; }
    x0 += ks1; x1 += ks2 + 4u;
#pragma unroll
    for (int i = 0; i < 4; ++i) { x0 += x1; x1 = rotl32(x1, rotA[i]); x1 ^= x0; }
    x0 += ks2; x1 += ks0 + 5u;
    o0 = x0; o1 = x1;
}

__global__ void sample_kernel(const float* __restrict__ mean_v,
                              const float* __restrict__ std_v,
                              float* __restrict__ out, long n) {
    long i = (long)blockIdx.x * blockDim.x + threadIdx.x;
    if (i >= n) return;
    unsigned h = (unsigned)(n >> 1);   // n is even (6.4M)
    unsigned c0, c1, o0, o1;
    int sel;
    if (i < (long)h) { c0 = (unsigned)i;       c1 = (unsigned)i + h; sel = 0; }
    else             { c0 = (unsigned)i - h;   c1 = (unsigned)i;     sel = 1; }
    threefry2x32(0u, 42u, c0, c1, o0, o1);
    unsigned bits = sel ? o1 : o0;
    // uniform in [lo, 1) with lo = nextafter(-1, 0), then sqrt(2)*erfinv
    float f = __uint_as_float((bits >> 9) | 0x3f800000u) - 1.0f;   // [0,1)
    const float lo = -0.99999994f;
    float u = f * (1.0f - lo) + lo;
    u = fmaxf(lo, u);
    float eps = 1.41421356237309515f * erfinvf(u);
    out[i] = mean_v[i] + eps * std_v[i];
}

// ---------------------------------------------------------------------------
extern "C" void kernel_launch(void* const* d_in, const int* in_sizes, int n_in,
                              void* d_out, int out_size, void* d_ws, size_t ws_size,
                              hipStream_t stream) {
    (void)n_in; (void)out_size; (void)ws_size;
    const float* x    = (const float*)d_in[0];
    const int*   esrc = (const int*)d_in[1];
    const int*   edst = (const int*)d_in[2];
    const float* P[2][7];
    for (int p = 0; p < 2; ++p)
        for (int j = 0; j < 7; ++j)
            P[p][j] = (const float*)d_in[3 + p * 7 + j];
    const float* gamma = (const float*)d_in[17];
    const float* beta  = (const float*)d_in[18];

    const int  n_nodes = in_sizes[0] / DDIM;   // 50000
    const int  n_edges = in_sizes[1];          // 1.6M
    const long ND = (long)n_nodes * DDIM;

    float* agg  = (float*)d_ws;
    float* tmp  = agg + ND;
    float* hbuf = tmp + ND;
    float* out    = (float*)d_out;
    float* mean_v = out + ND;
    float* std_v  = out + 2 * ND;

    dim3 blk(256);
    int blocks_elem = (int)((ND + 255) / 256);
    long scat_thr   = (long)n_edges * 32;
    int blocks_scat = (int)((scat_thr + 255) / 256);
    int tiles       = ((n_nodes + 15) / 16) * (DDIM / 16);   // 25000 (exact)
    int blocks_gemm = (tiles + 7) / 8;
    int blocks_ln   = (n_nodes + 7) / 8;

    for (int p = 0; p < 2; ++p) {
        const float* W1   = P[p][0];
        const float* b1   = P[p][1];
        const float* W2   = P[p][2];
        const float* b2   = P[p][3];
        const float* epsa = P[p][4];
        const float* Wout = P[p][5];
        const float* bout = P[p][6];

        const float* cur = x;
        for (int l = 0; l < 3; ++l) {
            scatter_init_kernel<<<blocks_elem, blk, 0, stream>>>(cur, agg, epsa, l, ND);
            edge_scatter_kernel<<<blocks_scat, blk, 0, stream>>>(cur, esrc, edst, agg, n_edges);
            gemm_bias_act_kernel<<<blocks_gemm, blk, 0, stream>>>(
                agg, W1 + (long)l * DDIM * DDIM, b1 + l * DDIM, tmp, n_nodes, 1);
            gemm_bias_act_kernel<<<blocks_gemm, blk, 0, stream>>>(
                tmp, W2 + (long)l * DDIM * DDIM, b2 + l * DDIM, hbuf, n_nodes, 0);
            cur = hbuf;
        }
        gemm_bias_act_kernel<<<blocks_gemm, blk, 0, stream>>>(
            cur, Wout, bout, tmp, n_nodes, 0);
        layernorm_kernel<<<blocks_ln, blk, 0, stream>>>(
            tmp, gamma, beta, (p == 0) ? mean_v : std_v, n_nodes);
    }

    sample_kernel<<<blocks_elem, blk, 0, stream>>>(mean_v, std_v, out, ND);
}